// _GCNLayer_5514738008947
// MI455X (gfx1250) — compile-verified
//
#include <hip/hip_runtime.h>
#include <hip/hip_bf16.h>

#define N_NODES 50000
#define N_EDGES 600000
#define D 128
#define LN_EPS 1e-5f

typedef __attribute__((ext_vector_type(2))) float v2f;
typedef __attribute__((ext_vector_type(8))) float v8f;

// ---------------------------------------------------------------------------
// Kernel 1: agg = h   (folds the "+ h" self term into the scatter base)
// ---------------------------------------------------------------------------
__global__ void gcn_init_agg(const float* __restrict__ h, float* __restrict__ agg) {
    int i = blockIdx.x * blockDim.x + threadIdx.x;
    if (i < N_NODES * (D / 4)) {
        ((float4*)agg)[i] = ((const float4*)h)[i];
    }
}

// ---------------------------------------------------------------------------
// Kernel 2: agg[dst] += h[src]  -- one wave per edge, float4 per lane,
// hardware global_atomic_add_f32 via unsafeAtomicAdd.
// ---------------------------------------------------------------------------
__global__ void gcn_scatter_add(const float* __restrict__ h,
                                const int* __restrict__ src,
                                const int* __restrict__ dst,
                                float* __restrict__ agg) {
    long g = (long)blockIdx.x * blockDim.x + threadIdx.x;
    int e = (int)(g >> 5);
    int lane = (int)(g & 31);
    if (e >= N_EDGES) return;
    int s = src[e];
    int d = dst[e];
    float4 v = ((const float4*)(h + (size_t)s * D))[lane];
    float* o = agg + (size_t)d * D + lane * 4;
    unsafeAtomicAdd(o + 0, v.x);
    unsafeAtomicAdd(o + 1, v.y);
    unsafeAtomicAdd(o + 2, v.z);
    unsafeAtomicAdd(o + 3, v.w);
}

// ---------------------------------------------------------------------------
// Kernel 3: y = relu(layernorm(agg @ W^T + b))
// 256 threads = 8 waves; each wave computes a 16-row x 128-col output stripe
// with 8 accumulator tiles of V_WMMA_F32_16X16X4_F32. W staged in LDS.
//
// f32 WMMA lane mapping (16x16 tile, K=4):
//   A : lane L -> row (L&15), K = 2*(L>>4) + {0,1}   (contiguous in x row)
//   B : lane L -> col (L&15), K = 2*(L>>4) + {0,1}   (contiguous in W[o][k])
//   C/D: lane L, VGPR v -> row v + 8*(L>>4), col (L&15)
// ---------------------------------------------------------------------------
__global__ __launch_bounds__(256) void gcn_gemm_ln_relu(
    const float* __restrict__ x,      // agg, [N, 128]
    const float* __restrict__ W,      // [128, 128] row-major [out][in]
    const float* __restrict__ bias,   // [128]
    const float* __restrict__ gamma,  // [128]
    const float* __restrict__ beta,   // [128]
    float* __restrict__ out)          // [N, 128]
{
    __shared__ float sW[D * D];  // 64 KB of the 320 KB WGP LDS

    {   // cooperative stage of W into LDS (reused by all 8 waves)
        const float4* Wv = (const float4*)W;
        float4* sWv = (float4*)sW;
        for (int i = threadIdx.x; i < (D * D) / 4; i += blockDim.x)
            sWv[i] = Wv[i];
    }
    __syncthreads();

    const int wave = threadIdx.x >> 5;
    const int lane = threadIdx.x & 31;
    const int rowBlock = blockIdx.x * 8 + wave;
    if (rowBlock * 16 >= N_NODES) return;   // wave-uniform: EXEC stays all-1s

    const int rowBase = rowBlock * 16;
    const int half = lane >> 4;   // 0: K%4 in {0,1} / rows 0..7 ; 1: {2,3} / rows 8..15
    const int l = lane & 15;

    v8f acc[8] = {};

    const float* xrow  = x + (size_t)(rowBase + l) * D + half * 2;
    const float* wbase = sW + half * 2;

    for (int kk = 0; kk < D; kk += 4) {
        v2f a = *(const v2f*)(xrow + kk);
#pragma unroll
        for (int t = 0; t < 8; ++t) {
            v2f bf = *(const v2f*)(wbase + (t * 16 + l) * D + kk);
            acc[t] = __builtin_amdgcn_wmma_f32_16x16x4_f32(
                false, a, false, bf, (short)0, acc[t], false, false);
        }
    }

    // per-lane column constants
    float bb[8], gg[8], be[8];
#pragma unroll
    for (int t = 0; t < 8; ++t) {
        int c = t * 16 + l;
        bb[t] = bias[c];
        gg[t] = gamma[c];
        be[t] = beta[c];
    }

    // add bias before LN statistics
#pragma unroll
    for (int t = 0; t < 8; ++t)
#pragma unroll
        for (int v = 0; v < 8; ++v)
            acc[t][v] += bb[t];

    // LayerNorm stats: row (v + 8*half) lives in VGPR v across the 16 lanes
    // of this half-wave x 8 tiles. Reduce across tiles then across 16 lanes.
    float mu[8], rs[8];
#pragma unroll
    for (int v = 0; v < 8; ++v) {
        float s = 0.0f, q = 0.0f;
#pragma unroll
        for (int t = 0; t < 8; ++t) {
            float xv = acc[t][v];
            s += xv;
            q += xv * xv;
        }
#pragma unroll
        for (int m = 1; m < 16; m <<= 1) {  // stays inside each 16-lane half
            s += __shfl_xor(s, m, 32);
            q += __shfl_xor(q, m, 32);
        }
        float mean = s * (1.0f / D);
        float var  = q * (1.0f / D) - mean * mean;
        mu[v] = mean;
        rs[v] = rsqrtf(var + LN_EPS);
    }

    // normalize + affine + ReLU + store
#pragma unroll
    for (int v = 0; v < 8; ++v) {
        float* orow = out + (size_t)(rowBase + v + half * 8) * D;
#pragma unroll
        for (int t = 0; t < 8; ++t) {
            float y = (acc[t][v] - mu[v]) * rs[v] * gg[t] + be[t];
            orow[t * 16 + l] = fmaxf(y, 0.0f);
        }
    }
}

// ---------------------------------------------------------------------------
extern "C" void kernel_launch(void* const* d_in, const int* in_sizes, int n_in,
                              void* d_out, int out_size, void* d_ws, size_t ws_size,
                              hipStream_t stream) {
    const float* h     = (const float*)d_in[0];
    const int*   edge  = (const int*)d_in[1];   // [2, E]: row0 = src, row1 = dst
    const float* W     = (const float*)d_in[2];
    const float* b     = (const float*)d_in[3];
    const float* gamma = (const float*)d_in[4];
    const float* beta  = (const float*)d_in[5];
    float* out = (float*)d_out;
    float* agg = (float*)d_ws;                  // N*128 f32 = 25.6 MB scratch

    const int* src = edge;
    const int* dst = edge + N_EDGES;

    // 1) agg = h
    {
        int n = N_NODES * (D / 4);
        gcn_init_agg<<<(n + 255) / 256, 256, 0, stream>>>(h, agg);
    }
    // 2) agg[dst] += h[src]
    {
        long total = (long)N_EDGES * 32;
        int blocks = (int)((total + 255) / 256);
        gcn_scatter_add<<<blocks, 256, 0, stream>>>(h, src, dst, agg);
    }
    // 3) out = relu(ln(agg @ W^T + b))
    {
        int rowBlocks = (N_NODES + 15) / 16;            // 3125
        int blocks = (rowBlocks + 7) / 8;               // 8 waves per block
        gcn_gemm_ln_relu<<<blocks, 256, 0, stream>>>(agg, W, b, gamma, beta, out);
    }
}